// SlotAttention_69346541961579
// MI455X (gfx1250) — compile-verified
//
#include <hip/hip_runtime.h>
#include <cstddef>

// ---------------- problem constants ----------------
#define Bb     64
#define Nn     4096
#define Dd     64
#define NSl    7
#define HIDD   128
#define NCHUNK 16          // token chunks per batch for the attention kernel
#define TPB    (Nn/NCHUNK) // 256 tokens per block
#define TILE   64          // tokens staged in LDS per pipelined step
#define LSTR   68          // LDS row stride (floats): 16B-aligned rows AND
                           // conflict-free column reads (bank = 4*row+col)
#define EPSf   1e-8f
#define LNEPS  1e-5f

typedef __attribute__((ext_vector_type(2))) float v2f;
typedef __attribute__((ext_vector_type(8))) float v8f;

// V_WMMA_F32_16X16X4_F32 : D(16x16,f32) = A(16x4,f32) x B(4x16,f32) + C
__device__ __forceinline__ v8f wmma4(v2f a, v2f b, v8f c) {
  return __builtin_amdgcn_wmma_f32_16x16x4_f32(false, a, false, b, (short)0, c,
                                               false, false);
}

__device__ __forceinline__ float sigm(float x) { return 1.f / (1.f + __expf(-x)); }

// Async-stage a 64x64 f32 tile (row-major, global) into LDS with row stride
// LSTR, using GLOBAL_LOAD_ASYNC_TO_LDS_B128 (ASYNCcnt path, VGPR-bypassing).
// 128 threads -> 8 rows x 16 x 16B chunks per pass, 8 passes, 8 async ops/wave.
__device__ __forceinline__ void async_tile_load(unsigned lds_byte_base,
                                                const float* __restrict__ gbase,
                                                int tid) {
  unsigned lofs = lds_byte_base + (unsigned)((tid >> 4) * (LSTR * 4) + (tid & 15) * 16);
  unsigned gofs = (unsigned)((tid >> 4) * 256 + (tid & 15) * 16);
  for (int j = 0; j < 8; ++j) {
    asm volatile("global_load_async_to_lds_b128 %0, %1, %2"
                 :
                 : "v"(lofs), "v"(gofs), "s"(gbase)
                 : "memory");
    lofs += 8u * LSTR * 4u;
    gofs += 8u * 256u;
  }
}

__device__ __forceinline__ void wait_async8() {
  asm volatile("s_wait_asynccnt 0x8" ::: "memory");
}
__device__ __forceinline__ void wait_async0() {
  asm volatile("s_wait_asynccnt 0x0" ::: "memory");
}

__device__ __forceinline__ unsigned lds_off(const void* p) {
  return (unsigned)(size_t)p;  // low 32 bits of generic ptr = LDS byte offset
}

// ---------------- kernel 1: slots init ----------------
__global__ void k_init_slots(const float* __restrict__ noise,
                             const float* __restrict__ mu,
                             const float* __restrict__ logsig,
                             float* __restrict__ slots) {
  int i = blockIdx.x * blockDim.x + threadIdx.x;
  if (i < Bb * NSl * Dd) {
    int d = i & (Dd - 1);
    slots[i] = mu[d] + __expf(logsig[d]) * noise[i];
  }
}

// ---------------- kernel 2: LN(inputs) -> k, v (f32 WMMA) ----------------
// 128 threads (4 waves), 64 rows per block. Weight staging runs as async
// LDS loads overlapped with the LayerNorm pass; GEMM via f32 WMMA.
__global__ __launch_bounds__(128) void k_ln_kv(
    const float* __restrict__ x, const float* __restrict__ Wk,
    const float* __restrict__ bk, const float* __restrict__ Wv,
    const float* __restrict__ bv, const float* __restrict__ g,
    const float* __restrict__ bb, float* __restrict__ kout,
    float* __restrict__ vout) {
  __shared__ __align__(16) float xln[64 * LSTR];
  __shared__ __align__(16) float wkl[64 * LSTR];
  __shared__ __align__(16) float wvl[64 * LSTR];

  const int tid = threadIdx.x;
  const int row0 = blockIdx.x * 64;

  // async weight staging (16 ops/wave in flight) overlapped with LN below
  async_tile_load(lds_off(wkl), Wk, tid);
  async_tile_load(lds_off(wvl), Wv, tid);

  // LayerNorm: 2 threads per row, 32 elements each; combine via lane shuffle.
  {
    int r = tid >> 1, half = tid & 1;
    const float* xr = x + (size_t)(row0 + r) * Dd + half * 32;
    float vals[32];
    float s = 0.f, s2 = 0.f;
    for (int c = 0; c < 32; ++c) {
      float t = xr[c];
      vals[c] = t;
      s += t;
      s2 += t * t;
    }
    s += __shfl_xor(s, 1, 32);
    s2 += __shfl_xor(s2, 1, 32);
    float m = s * (1.f / 64.f);
    float var = s2 * (1.f / 64.f) - m * m;
    float rstd = rsqrtf(var + LNEPS);
    for (int c = 0; c < 32; ++c) {
      int cc = half * 32 + c;
      xln[r * LSTR + cc] = (vals[c] - m) * rstd * g[cc] + bb[cc];
    }
  }
  wait_async0();  // weights landed
  __syncthreads();

  const int wave = tid >> 5, lane = tid & 31;
  const int m = lane & 15, kh = lane >> 4;
  const int mbase = wave * 16;

  v8f ck[4], cv[4];
  for (int n2 = 0; n2 < 4; ++n2) {
    ck[n2] = (v8f){0.f, 0.f, 0.f, 0.f, 0.f, 0.f, 0.f, 0.f};
    cv[n2] = (v8f){0.f, 0.f, 0.f, 0.f, 0.f, 0.f, 0.f, 0.f};
  }
  for (int kk = 0; kk < 16; ++kk) {
    int c0 = kk * 4 + kh * 2;
    v2f a;
    a.x = xln[(mbase + m) * LSTR + c0];
    a.y = xln[(mbase + m) * LSTR + c0 + 1];
    for (int n2 = 0; n2 < 4; ++n2) {
      int d = n2 * 16 + m;  // B lane = output column
      v2f b1, b2;
      b1.x = wkl[d * LSTR + c0];
      b1.y = wkl[d * LSTR + c0 + 1];
      b2.x = wvl[d * LSTR + c0];
      b2.y = wvl[d * LSTR + c0 + 1];
      ck[n2] = wmma4(a, b1, ck[n2]);
      cv[n2] = wmma4(a, b2, cv[n2]);
    }
  }
  // D layout: VGPR r -> M=r (lanes 0-15) / M=8+r (lanes 16-31); lane&15 -> N
  const int mofs = kh * 8, nl = m;
  for (int n2 = 0; n2 < 4; ++n2) {
    int col = n2 * 16 + nl;
    for (int r = 0; r < 8; ++r) {
      size_t row = (size_t)row0 + mbase + mofs + r;
      kout[row * Dd + col] = ck[n2][r] + bk[col];
      vout[row * Dd + col] = cv[n2][r] + bv[col];
    }
  }
}

// ---------------- kernel 3: LN(slots) -> q (tiny) ----------------
__global__ __launch_bounds__(64) void k_slots_q(
    const float* __restrict__ slots, const float* __restrict__ Wq,
    const float* __restrict__ bq, const float* __restrict__ g,
    const float* __restrict__ bb, float* __restrict__ qout) {
  __shared__ float sl[NSl][Dd];
  __shared__ float sln[NSl][Dd];
  __shared__ float st[NSl][2];
  const int b = blockIdx.x, t = threadIdx.x;
  for (int e = t; e < NSl * Dd; e += 64) sl[e >> 6][e & 63] = slots[b * NSl * Dd + e];
  __syncthreads();
  if (t < NSl) {
    float s = 0.f, s2 = 0.f;
    for (int c = 0; c < Dd; ++c) {
      float v = sl[t][c];
      s += v;
      s2 += v * v;
    }
    float m = s * (1.f / 64.f);
    st[t][0] = m;
    st[t][1] = rsqrtf(s2 * (1.f / 64.f) - m * m + LNEPS);
  }
  __syncthreads();
  for (int i = 0; i < NSl; ++i)
    sln[i][t] = (sl[i][t] - st[i][0]) * st[i][1] * g[t] + bb[t];
  __syncthreads();
  // q[i][t] = sum_c sln[i][c] * Wq[t][c] + bq[t]; pad rows 7..15 with zeros
  const float* w = Wq + t * Dd;
  for (int i = 0; i < NSl; ++i) {
    float acc = bq[t];
    for (int c = 0; c < Dd; ++c) acc += sln[i][c] * w[c];
    qout[(size_t)b * 16 * Dd + i * Dd + t] = acc;
  }
  for (int i = NSl; i < 16; ++i) qout[(size_t)b * 16 * Dd + i * Dd + t] = 0.f;
}

// ---------------- kernel 4: fused dots/softmax/updates partials ----------------
// grid (B, NCHUNK), 128 threads (4 waves), 4 pipelined 64-token steps.
// k/v tiles staged by GLOBAL_LOAD_ASYNC_TO_LDS_B128 (ASYNCcnt), double-phase
// pipelined: v(s) prefetch overlaps dots WMMA on k(s); k(s+1) prefetch
// overlaps the p x v updates WMMA on v(s). Slot logits land in accumulator
// VGPRs 0..6 of lanes 0..15 -> per-lane softmax over slots, +EPS; token
// renormalization folded into per-slot scalar Z.
__global__ __launch_bounds__(128) void k_attn(
    const float* __restrict__ qg, const float* __restrict__ kg,
    const float* __restrict__ vg, float* __restrict__ Upart,
    float* __restrict__ Zpart) {
  __shared__ __align__(16) float qs[16 * LSTR];
  __shared__ __align__(16) float bufk[TILE * LSTR];
  __shared__ __align__(16) float bufv[TILE * LSTR];
  __shared__ float pl[16 * LSTR];  // p, rows 7..15 stay zero
  __shared__ float ured[4][NSl][Dd];
  __shared__ float zred[4][NSl][16];

  const int b = blockIdx.x, ch = blockIdx.y;
  const int tid = threadIdx.x;
  const int wave = tid >> 5, lane = tid & 31;
  const int m = lane & 15, kh = lane >> 4;

  const int tok0 = ch * TPB;
  const float* kbase = kg + ((size_t)b * Nn + tok0) * Dd;
  const float* vbase = vg + ((size_t)b * Nn + tok0) * Dd;
  const unsigned lk = lds_off(bufk), lv = lds_off(bufv);

  async_tile_load(lk, kbase, tid);  // prefetch k(0)

  for (int e = tid; e < 16 * 64; e += 128)
    qs[(e >> 6) * LSTR + (e & 63)] = qg[(size_t)b * 16 * Dd + e];
  for (int e = tid; e < 16 * LSTR; e += 128) pl[e] = 0.f;

  v8f uacc[4];
  for (int d2 = 0; d2 < 4; ++d2)
    uacc[d2] = (v8f){0.f, 0.f, 0.f, 0.f, 0.f, 0.f, 0.f, 0.f};
  float zacc[NSl] = {0.f, 0.f, 0.f, 0.f, 0.f, 0.f, 0.f};

  for (int step = 0; step < TPB / TILE; ++step) {
    async_tile_load(lv, vbase + (size_t)step * TILE * Dd, tid);  // prefetch v(s)
    wait_async8();  // k(s) landed (in-order completion), v(s) still in flight
    __syncthreads();

    // --- dots on k tile: this wave's 16 tokens ---
    {
      const int ncol0 = wave * 16;
      v8f c8 = (v8f){0.f, 0.f, 0.f, 0.f, 0.f, 0.f, 0.f, 0.f};
      for (int kk = 0; kk < 16; ++kk) {
        int c0 = kk * 4 + kh * 2;
        v2f a, bv_;
        a.x = qs[m * LSTR + c0];
        a.y = qs[m * LSTR + c0 + 1];
        int n = ncol0 + m;  // B lane = token column; B = k^T
        bv_.x = bufk[n * LSTR + c0];
        bv_.y = bufk[n * LSTR + c0 + 1];
        c8 = wmma4(a, bv_, c8);
      }
      if (lane < 16) {  // lanes 0..15 hold slot rows 0..7 of one token each
        float d[NSl];
        float mx = -3.4e38f;
        for (int i = 0; i < NSl; ++i) {
          d[i] = c8[i] * 0.125f;  // scale = D^-0.5
          mx = fmaxf(mx, d[i]);
        }
        float sum = 0.f;
        for (int i = 0; i < NSl; ++i) {
          d[i] = __expf(d[i] - mx);
          sum += d[i];
        }
        float inv = 1.f / sum;
        for (int i = 0; i < NSl; ++i) {
          float p = d[i] * inv + EPSf;
          zacc[i] += p;
          pl[i * LSTR + ncol0 + lane] = p;
        }
      }
    }
    __syncthreads();  // all waves done reading bufk

    if (step < TPB / TILE - 1) {
      async_tile_load(lk, kbase + (size_t)(step + 1) * TILE * Dd, tid);
      wait_async8();  // v(s) landed, k(s+1) in flight
    } else {
      wait_async0();  // v(s) landed
    }
    __syncthreads();

    // --- U += p x v over this wave's 16 tokens (K dim) ---
    for (int kk = 0; kk < 4; ++kk) {
      int t0 = wave * 16 + kk * 4 + kh * 2;
      v2f a;
      a.x = pl[m * LSTR + t0];
      a.y = pl[m * LSTR + t0 + 1];
      for (int d2 = 0; d2 < 4; ++d2) {
        int d = d2 * 16 + m;
        v2f bv_;
        bv_.x = bufv[t0 * LSTR + d];
        bv_.y = bufv[(t0 + 1) * LSTR + d];
        uacc[d2] = wmma4(a, bv_, uacc[d2]);
      }
    }
    __syncthreads();  // all waves done reading bufv before next v prefetch
  }

  // --- block reduce 4 waves, emit deterministic per-chunk partials ---
  if (lane < 16) {
    for (int d2 = 0; d2 < 4; ++d2)
      for (int i = 0; i < NSl; ++i) ured[wave][i][d2 * 16 + lane] = uacc[d2][i];
    for (int i = 0; i < NSl; ++i) zred[wave][i][lane] = zacc[i];
  }
  __syncthreads();
  for (int e = tid; e < NSl * Dd; e += 128) {
    int i = e >> 6, d = e & 63;
    Upart[((size_t)b * NCHUNK + ch) * (NSl * Dd) + e] =
        ured[0][i][d] + ured[1][i][d] + ured[2][i][d] + ured[3][i][d];
  }
  if (tid < NSl) {
    float s = 0.f;
    for (int w = 0; w < 4; ++w)
      for (int l = 0; l < 16; ++l) s += zred[w][tid][l];
    Zpart[((size_t)b * NCHUNK + ch) * NSl + tid] = s;
  }
}

// ---------------- kernel 5: reduce + GRU + LN + MLP -> new slots ----------------
__global__ __launch_bounds__(128) void k_update(
    float* __restrict__ slots, const float* __restrict__ Upart,
    const float* __restrict__ Zpart, const float* __restrict__ W_ih,
    const float* __restrict__ b_ih, const float* __restrict__ W_hh,
    const float* __restrict__ b_hh, const float* __restrict__ W1,
    const float* __restrict__ b1, const float* __restrict__ W2,
    const float* __restrict__ b2, const float* __restrict__ gm,
    const float* __restrict__ bm) {
  __shared__ float upd[NSl][Dd];
  __shared__ float sprev[NSl][Dd];
  __shared__ float gx[NSl][3 * Dd];
  __shared__ float gh[NSl][3 * Dd];
  __shared__ float snew[NSl][Dd];
  __shared__ float hln[NSl][Dd];
  __shared__ float hid[NSl][HIDD];
  __shared__ float st[NSl][2];
  const int b = blockIdx.x, t = threadIdx.x;

  if (t < NSl) {  // token-renorm denominator Z_i
    float z = 0.f;
    for (int ch = 0; ch < NCHUNK; ++ch)
      z += Zpart[((size_t)b * NCHUNK + ch) * NSl + t];
    st[t][0] = z;
  }
  for (int e = t; e < NSl * Dd; e += 128) {
    float s = 0.f;
    for (int ch = 0; ch < NCHUNK; ++ch)
      s += Upart[((size_t)b * NCHUNK + ch) * (NSl * Dd) + e];
    upd[e >> 6][e & 63] = s;
    sprev[e >> 6][e & 63] = slots[(size_t)b * NSl * Dd + e];
  }
  __syncthreads();
  for (int e = t; e < NSl * Dd; e += 128) upd[e >> 6][e & 63] /= st[e >> 6][0];
  __syncthreads();
  // GRU gate pre-activations
  for (int e = t; e < NSl * 3 * Dd; e += 128) {
    int i = e / (3 * Dd), j = e % (3 * Dd);
    const float* wi = W_ih + j * Dd;
    const float* wh = W_hh + j * Dd;
    float a1 = b_ih[j], a2 = b_hh[j];
    for (int c = 0; c < Dd; ++c) {
      a1 += upd[i][c] * wi[c];
      a2 += sprev[i][c] * wh[c];
    }
    gx[i][j] = a1;
    gh[i][j] = a2;
  }
  __syncthreads();
  for (int e = t; e < NSl * Dd; e += 128) {
    int i = e >> 6, d = e & 63;
    float r = sigm(gx[i][d] + gh[i][d]);
    float z = sigm(gx[i][Dd + d] + gh[i][Dd + d]);
    float n = tanhf(gx[i][2 * Dd + d] + r * gh[i][2 * Dd + d]);
    snew[i][d] = (1.f - z) * n + z * sprev[i][d];
  }
  __syncthreads();
  if (t < NSl) {
    float s = 0.f, s2 = 0.f;
    for (int c = 0; c < Dd; ++c) {
      float v = snew[t][c];
      s += v;
      s2 += v * v;
    }
    float m = s * (1.f / 64.f);
    st[t][0] = m;
    st[t][1] = rsqrtf(s2 * (1.f / 64.f) - m * m + LNEPS);
  }
  __syncthreads();
  for (int e = t; e < NSl * Dd; e += 128) {
    int i = e >> 6, d = e & 63;
    hln[i][d] = (snew[i][d] - st[i][0]) * st[i][1] * gm[d] + bm[d];
  }
  __syncthreads();
  for (int e = t; e < NSl * HIDD; e += 128) {
    int i = e >> 7, j = e & 127;
    const float* w = W1 + j * Dd;
    float a = b1[j];
    for (int c = 0; c < Dd; ++c) a += hln[i][c] * w[c];
    hid[i][j] = fmaxf(a, 0.f);
  }
  __syncthreads();
  for (int e = t; e < NSl * Dd; e += 128) {
    int i = e >> 6, d = e & 63;
    const float* w = W2 + d * HIDD;
    float a = b2[d];
    for (int j = 0; j < HIDD; ++j) a += hid[i][j] * w[j];
    slots[(size_t)b * NSl * Dd + e] = snew[i][d] + a;
  }
}

// ---------------- host launcher ----------------
extern "C" void kernel_launch(void* const* d_in, const int* in_sizes, int n_in,
                              void* d_out, int out_size, void* d_ws,
                              size_t ws_size, hipStream_t stream) {
  const float* inputs = (const float*)d_in[0];
  const float* noise = (const float*)d_in[1];
  const float* slots_mu = (const float*)d_in[2];
  const float* slots_ls = (const float*)d_in[3];
  const float* Wq = (const float*)d_in[4];
  const float* bq = (const float*)d_in[5];
  const float* Wk = (const float*)d_in[6];
  const float* bk = (const float*)d_in[7];
  const float* Wv = (const float*)d_in[8];
  const float* bv = (const float*)d_in[9];
  const float* W_ih = (const float*)d_in[10];
  const float* b_ih = (const float*)d_in[11];
  const float* W_hh = (const float*)d_in[12];
  const float* b_hh = (const float*)d_in[13];
  const float* W1 = (const float*)d_in[14];
  const float* b1 = (const float*)d_in[15];
  const float* W2 = (const float*)d_in[16];
  const float* b2 = (const float*)d_in[17];
  const float* ln_in_g = (const float*)d_in[18];
  const float* ln_in_b = (const float*)d_in[19];
  const float* ln_s_g = (const float*)d_in[20];
  const float* ln_s_b = (const float*)d_in[21];
  const float* ln_m_g = (const float*)d_in[22];
  const float* ln_m_b = (const float*)d_in[23];

  float* ws = (float*)d_ws;
  float* slots = ws;                              // 28672
  float* qg = slots + Bb * NSl * Dd;              // 65536
  float* kg = qg + (size_t)Bb * 16 * Dd;          // 16777216
  float* vg = kg + (size_t)Bb * Nn * Dd;          // 16777216
  float* Upart = vg + (size_t)Bb * Nn * Dd;       // 458752
  float* Zpart = Upart + (size_t)Bb * NCHUNK * NSl * Dd;  // 7168

  k_init_slots<<<(Bb * NSl * Dd + 255) / 256, 256, 0, stream>>>(
      noise, slots_mu, slots_ls, slots);
  k_ln_kv<<<(Bb * Nn) / 64, 128, 0, stream>>>(inputs, Wk, bk, Wv, bv, ln_in_g,
                                              ln_in_b, kg, vg);
  for (int it = 0; it < 3; ++it) {
    k_slots_q<<<Bb, 64, 0, stream>>>(slots, Wq, bq, ln_s_g, ln_s_b, qg);
    dim3 ag(Bb, NCHUNK);
    k_attn<<<ag, 128, 0, stream>>>(qg, kg, vg, Upart, Zpart);
    k_update<<<Bb, 128, 0, stream>>>(slots, Upart, Zpart, W_ih, b_ih, W_hh,
                                     b_hh, W1, b1, W2, b2, ln_m_g, ln_m_b);
  }
  hipMemcpyAsync(d_out, slots, (size_t)Bb * NSl * Dd * sizeof(float),
                 hipMemcpyDeviceToDevice, stream);
}